// DynamicWeightedLoss_51479478010334
// MI455X (gfx1250) — compile-verified
//
#include <hip/hip_runtime.h>
#include <cstdint>
#include <cstddef>

// Problem constants (match reference)
#define BB      16
#define CCH     4
#define HH      512
#define WW      512
#define HWP     (HH * WW)          // 262144 = 2^18
#define NPIX    (BB * HWP)         // 4,194,304
#define ALPHA_C 1.0f
#define BETA_C  0.7f
#define UPPER_C 1000.0f
#define LOWER_C 200.0f
#define WUP_C   2.5f
#define WLO_C   20.0f
#define SMOOTH_C 1.0f

#define THREADS 256

typedef float v4f __attribute__((ext_vector_type(4)));   // clang vector: OK for nontemporal builtin

// Accumulator layout (floats): [0]=wce_sum [1]=focal_sum
// [2..65]=prob_sum[b*4+c] [66..129]=inter[b*4+c] [130..193]=onehot_cnt[b*4+c]
#define ACC_PS 2
#define ACC_IN 66
#define ACC_CN 130
#define ACC_TOTAL 194

// ---------------------------------------------------------------------------
// CDNA5 async global->LDS helpers (gfx1250 ASYNC path, tracked by ASYNCcnt)
// ---------------------------------------------------------------------------
__device__ __forceinline__ void async_load_b128_to_lds(uint32_t lds_off, uint64_t gaddr) {
    asm volatile("global_load_async_to_lds_b128 %0, %1, off"
                 :: "v"(lds_off), "v"(gaddr) : "memory");
}
__device__ __forceinline__ void wait_asynccnt0() {
    asm volatile("s_wait_asynccnt 0" ::: "memory");
}

// ---------------------------------------------------------------------------
// Lock-free union-find (ECL-CC style) for 4-connected components of tgt==2
// ---------------------------------------------------------------------------
__device__ __forceinline__ int uf_find(int* __restrict__ p, int i) {
    int q = p[i];
    while (q != i) { i = q; q = p[i]; }
    return i;
}

__device__ __forceinline__ void uf_union(int* __restrict__ p, int a, int b) {
    while (true) {
        a = uf_find(p, a);
        b = uf_find(p, b);
        if (a == b) return;
        if (a < b) { int old = atomicMin(&p[b], a); if (old == b) return; b = old; }
        else       { int old = atomicMin(&p[a], b); if (old == a) return; a = old; }
    }
}

__global__ void k_init(int* __restrict__ parent, int* __restrict__ count,
                       float* __restrict__ accum) {
    int g = blockIdx.x * THREADS + threadIdx.x;     // exact cover of NPIX
    parent[g] = g;
    count[g]  = 0;
    if (g < ACC_TOTAL) accum[g] = 0.0f;
}

__global__ void k_merge(int* __restrict__ parent, const int* __restrict__ tgt) {
    int g = blockIdx.x * THREADS + threadIdx.x;
    if (tgt[g] != 2) return;
    int x = g & (WW - 1);
    int y = (g >> 9) & (HH - 1);                    // row within image
    if (x > 0  && tgt[g - 1]  == 2) uf_union(parent, g, g - 1);
    if (y > 0  && tgt[g - WW] == 2) uf_union(parent, g, g - WW);
}

// Flatten labels to roots and count component areas. wave32-aggregated
// atomics: lanes cover 32 contiguous pixels of one row; only the head lane
// of each equal-root run issues a single atomicAdd of the run length.
__global__ void k_flatten_count(int* __restrict__ parent, int* __restrict__ count,
                                const int* __restrict__ tgt) {
    int g = blockIdx.x * THREADS + threadIdx.x;
    bool msk = (tgt[g] == 2);
    int r = -1;
    if (msk) {
        r = uf_find(parent, g);
        parent[g] = r;
    }
    int lane  = threadIdx.x & 31;
    int rprev = __shfl_up(r, 1);
    bool boundary = (lane == 0) || (rprev != r);
    unsigned long long B = __ballot(boundary);      // wave32: bits [31:0]
    if (msk && boundary) {
        unsigned long long rest = B >> (lane + 1);  // lane+1 <= 32, 64-bit shift OK
        int len = rest ? __ffsll(rest) : (32 - lane);
        atomicAdd(&count[r], len);
    }
}

// ---------------------------------------------------------------------------
// Fused main pass: softmax/CE + dynamic weight + focal + dice accumulators.
// One block = 256 threads * 4 pixels = 1024 pixels, always inside one image.
// ---------------------------------------------------------------------------
__global__ void k_main(const float* __restrict__ outp, const float* __restrict__ cw,
                       const int*  __restrict__ tgt,  const int* __restrict__ parent,
                       const int*  __restrict__ count, float* __restrict__ accum) {
    __shared__ int   s_tgt[THREADS * 4];
    __shared__ float s_acc[14];

    const int tid = threadIdx.x;
    if (tid < 14) s_acc[tid] = 0.0f;
    __syncthreads();

    const int g4 = blockIdx.x * THREADS + tid;   // group-of-4 pixel index
    const int gp = g4 << 2;                      // global pixel base
    const int b  = gp >> 18;                     // image (uniform per block)
    const int pi = gp & (HWP - 1);               // pixel within image

    // Stage this thread's 4 targets via the CDNA5 async-copy engine.
    uint32_t lds_off = (uint32_t)(uintptr_t)(&s_tgt[tid * 4]);
    async_load_b128_to_lds(lds_off, (uint64_t)(uintptr_t)(tgt + gp));

    // Overlap: stream the 4 channel-plane float4's (read-once -> nontemporal)
    // while the async copy runs.
    const float* base = outp + (size_t)b * (4 * HWP) + pi;
    v4f c0 = __builtin_nontemporal_load((const v4f*)(base + 0 * HWP));
    v4f c1 = __builtin_nontemporal_load((const v4f*)(base + 1 * HWP));
    v4f c2 = __builtin_nontemporal_load((const v4f*)(base + 2 * HWP));
    v4f c3 = __builtin_nontemporal_load((const v4f*)(base + 3 * HWP));

    const float cw0 = cw[0], cw1 = cw[1], cw2 = cw[2], cw3 = cw[3];

    wait_asynccnt0();
    int4 tv = *(const int4*)(&s_tgt[tid * 4]);

    float wce = 0.f, foc = 0.f;
    float ps0 = 0.f, ps1 = 0.f, ps2 = 0.f, ps3 = 0.f;
    float in0 = 0.f, in1 = 0.f, in2 = 0.f, in3 = 0.f;
    float cn0 = 0.f, cn1 = 0.f, cn2 = 0.f, cn3 = 0.f;

    auto pixel = [&](float a0, float a1, float a2, float a3, int t, int pix) {
        float m  = fmaxf(fmaxf(a0, a1), fmaxf(a2, a3));
        float e0 = expf(a0 - m), e1 = expf(a1 - m), e2 = expf(a2 - m), e3 = expf(a3 - m);
        float s  = e0 + e1 + e2 + e3;
        float inv = 1.0f / s;
        float lse = m + logf(s);
        float xt  = (t == 0) ? a0 : (t == 1) ? a1 : (t == 2) ? a2 : a3;
        float cwt = (t == 0) ? cw0 : (t == 1) ? cw1 : (t == 2) ? cw2 : cw3;
        float ce  = cwt * (lse - xt);

        float dynw = 1.0f;
        if (t == 2) {
            int   r    = parent[pix];
            float area = (float)count[r];
            float interp = WUP_C + (WLO_C - WUP_C) * (UPPER_C - area) * (1.0f / (UPPER_C - LOWER_C));
            dynw = (area > UPPER_C) ? WUP_C : ((area < LOWER_C) ? WLO_C : interp);
        }
        wce += ce * dynw;

        float pt  = expf(-ce);
        float omp = 1.0f - pt;
        foc += ALPHA_C * omp * omp * ce;

        ps0 += e0 * inv; ps1 += e1 * inv; ps2 += e2 * inv; ps3 += e3 * inv;
        float ptc = ((t == 0) ? e0 : (t == 1) ? e1 : (t == 2) ? e2 : e3) * inv;
        if      (t == 0) { in0 += ptc; cn0 += 1.0f; }
        else if (t == 1) { in1 += ptc; cn1 += 1.0f; }
        else if (t == 2) { in2 += ptc; cn2 += 1.0f; }
        else             { in3 += ptc; cn3 += 1.0f; }
    };

    pixel(c0.x, c1.x, c2.x, c3.x, tv.x, gp + 0);
    pixel(c0.y, c1.y, c2.y, c3.y, tv.y, gp + 1);
    pixel(c0.z, c1.z, c2.z, c3.z, tv.z, gp + 2);
    pixel(c0.w, c1.w, c2.w, c3.w, tv.w, gp + 3);

    // wave32 shuffle reduction -> 14 LDS bins -> 14 global atomics per block
    float vals[14] = {wce, foc, ps0, ps1, ps2, ps3,
                      in0, in1, in2, in3, cn0, cn1, cn2, cn3};
#pragma unroll
    for (int k = 0; k < 14; ++k) {
        float v = vals[k];
#pragma unroll
        for (int off = 16; off > 0; off >>= 1) v += __shfl_down(v, off);
        if ((tid & 31) == 0) atomicAdd(&s_acc[k], v);
    }
    __syncthreads();
    if (tid < 14) {
        int k = tid;
        int dst = (k == 0) ? 0
                : (k == 1) ? 1
                : (k < 6)  ? (ACC_PS + b * 4 + (k - 2))
                : (k < 10) ? (ACC_IN + b * 4 + (k - 6))
                :            (ACC_CN + b * 4 + (k - 10));
        atomicAdd(&accum[dst], s_acc[k]);
    }
}

__global__ void k_finalize(const float* __restrict__ accum, float* __restrict__ out) {
    __shared__ float sh[64];
    int i = threadIdx.x;
    float d = 0.0f;
    if (i < 64) {
        float inter = accum[ACC_IN + i];
        float ps    = accum[ACC_PS + i];
        float cn    = accum[ACC_CN + i];
        d = (2.0f * inter + SMOOTH_C) / (ps + cn + SMOOTH_C);
    }
    sh[i] = d;
    __syncthreads();
    for (int s = 32; s > 0; s >>= 1) { if (i < s) sh[i] += sh[i + s]; __syncthreads(); }
    if (i == 0) {
        float dice_loss = 1.0f - sh[0] * (1.0f / 64.0f);
        float invN = 1.0f / (float)NPIX;
        float wce = accum[0] * invN;
        float foc = accum[1] * invN;
        out[0] = BETA_C * wce + (1.0f - BETA_C) * dice_loss + foc;
    }
}

// ---------------------------------------------------------------------------
extern "C" void kernel_launch(void* const* d_in, const int* in_sizes, int n_in,
                              void* d_out, int out_size, void* d_ws, size_t ws_size,
                              hipStream_t stream) {
    const float* outputs = (const float*)d_in[0];
    const float* cw      = (const float*)d_in[1];
    const int*   targets = (const int*)d_in[2];

    int*   parent = (int*)d_ws;
    int*   count  = parent + NPIX;
    float* accum  = (float*)(count + NPIX);
    float* out    = (float*)d_out;

    dim3 blk(THREADS);
    k_init         <<<NPIX / THREADS,        blk, 0, stream>>>(parent, count, accum);
    k_merge        <<<NPIX / THREADS,        blk, 0, stream>>>(parent, targets);
    k_flatten_count<<<NPIX / THREADS,        blk, 0, stream>>>(parent, count, targets);
    k_main         <<<NPIX / (THREADS * 4),  blk, 0, stream>>>(outputs, cw, targets,
                                                               parent, count, accum);
    k_finalize     <<<1, 64, 0, stream>>>(accum, out);
}